// TopKPool_40673340293785
// MI455X (gfx1250) — compile-verified
//
#include <hip/hip_runtime.h>
#include <hip/hip_bf16.h>

// ---------------------------------------------------------------------------
// TopK pooling (SAGPool): y = X p/||p||; top-8192 of y; X_pooled = (X*tanh(y))[idx];
// A_pooled = A[idx][:,idx].  N=16384, F=256, K=8192.
// Dominant cost: A_pooled gather (~768 MB traffic -> ~33us @ 23.3 TB/s).
// Matrix math (X.p) done with native f32 WMMA: V_WMMA_F32_16X16X4_F32.
// ---------------------------------------------------------------------------

#define TK_N 16384
#define TK_F 256
#define TK_K 8192

typedef __attribute__((ext_vector_type(2))) float v2f;
typedef __attribute__((ext_vector_type(8))) float v8f;

// ---- K0: ||p|| -------------------------------------------------------------
__global__ void tk_pnorm_kernel(const float* __restrict__ p, float* __restrict__ pn) {
    __shared__ float red[TK_F];
    int t = threadIdx.x;
    float v = p[t];
    red[t] = v * v;
    __syncthreads();
    for (int s = TK_F / 2; s > 0; s >>= 1) {
        if (t < s) red[t] += red[t + s];
        __syncthreads();
    }
    if (t == 0) pn[0] = sqrtf(red[0]);
}

// ---- K1: scores y = X p / ||p|| via V_WMMA_F32_16X16X4_F32 -----------------
// One wave computes 16 rows. A = 16x4 tile of X (lanes 0-15: K={0,1},
// lanes 16-31: K={2,3}, 2 VGPRs). B = 4x16 with p replicated in all 16
// columns (same K striping). D: lane 0 col0 holds M=0..7, lane 16 holds M=8..15.
__global__ void tk_scores_kernel(const float* __restrict__ X,
                                 const float* __restrict__ p,
                                 const float* __restrict__ pn,
                                 float* __restrict__ y,
                                 unsigned* __restrict__ keys) {
    int wave = threadIdx.x >> 5;
    int lane = threadIdx.x & 31;
    int row0 = blockIdx.x * 64 + wave * 16;
    int M    = lane & 15;
    int koff = (lane >> 4) * 2;

    const float* xrow = X + (size_t)(row0 + M) * TK_F + koff;
    const float* pptr = p + koff;

    v8f c = {};
    for (int kb = 0; kb < TK_F; kb += 4) {
        v2f a = *(const v2f*)(xrow + kb);
        v2f b = *(const v2f*)(pptr + kb);
        // 8 args: (neg_a, A, neg_b, B, c_mod, C, reuse_a, reuse_b)
        c = __builtin_amdgcn_wmma_f32_16x16x4_f32(
                false, a, false, b, (short)0, c, false, false);
    }

    float inv = 1.0f / pn[0];
    if ((lane & 15) == 0) {            // lane 0 -> M=0..7, lane 16 -> M=8..15
        int mbase = row0 + (lane >> 4) * 8;
        #pragma unroll
        for (int v = 0; v < 8; ++v) {
            float yv = c[v] * inv;
            y[mbase + v] = yv;
            unsigned u = __float_as_uint(yv);
            keys[mbase + v] = (u & 0x80000000u) ? ~u : (u | 0x80000000u);
        }
    }
}

// ---- K2: exact top-K threshold via 4-round radix histogram + stable compact
// Single block, 1024 threads. Deterministic: ties kept by lowest index
// (matches jax.lax.top_k), output idx[] is ascending by construction.
__global__ void tk_select_kernel(const unsigned* __restrict__ keys,
                                 int* __restrict__ idx) {
    __shared__ unsigned bins[256];
    __shared__ unsigned cg[1024];
    __shared__ unsigned ce[1024];
    __shared__ unsigned s_prefix, s_krem;

    int t = threadIdx.x;
    if (t == 0) { s_prefix = 0u; s_krem = TK_K; }
    __syncthreads();

    unsigned prefmask = 0u;
    for (int shift = 24; shift >= 0; shift -= 8) {
        if (t < 256) bins[t] = 0u;
        __syncthreads();
        unsigned pref = s_prefix;
        for (int i = t; i < TK_N; i += 1024) {
            unsigned k = keys[i];
            if ((k & prefmask) == pref)
                atomicAdd(&bins[(k >> shift) & 0xFFu], 1u);
        }
        __syncthreads();
        if (t == 0) {
            unsigned krem = s_krem, cum = 0u;
            int chosen = 0;
            for (int b = 255; b >= 0; --b) {
                unsigned cb = bins[b];
                if (cum + cb >= krem) { chosen = b; break; }
                cum += cb;
            }
            s_prefix = pref | ((unsigned)chosen << shift);
            s_krem   = krem - cum;      // rank remaining within chosen bin
        }
        prefmask |= (0xFFu << shift);
        __syncthreads();
    }

    unsigned T    = s_prefix;   // exact K-th largest key
    unsigned ties = s_krem;     // # of keys == T to keep (lowest indices first)

    // Stable compaction: 16 contiguous elements per thread.
    int base = t * 16;
    unsigned gt_c = 0u, eq_c = 0u;
    for (int i = base; i < base + 16; ++i) {
        unsigned k = keys[i];
        gt_c += (k > T);
        eq_c += (k == T);
    }
    cg[t] = gt_c; ce[t] = eq_c;
    __syncthreads();
    if (t == 0) {               // serial exclusive scan over 1024 chunks
        unsigned ag = 0u, ae = 0u;
        for (int i = 0; i < 1024; ++i) {
            unsigned g = cg[i], e = ce[i];
            cg[i] = ag; ce[i] = ae;
            ag += g; ae += e;
        }
    }
    __syncthreads();
    unsigned g = cg[t], e = ce[t];
    for (int i = base; i < base + 16; ++i) {
        unsigned k = keys[i];
        if (k > T) {
            unsigned pos = g + (e < ties ? e : ties);
            idx[pos] = i;
            ++g;
        } else if (k == T) {
            if (e < ties) idx[g + e] = i;
            ++e;
        }
    }
}

// ---- K3: X_pooled[k,:] = X[idx[k],:] * tanh(y[idx[k]]) ---------------------
__global__ void tk_xpool_kernel(const float* __restrict__ X,
                                const float* __restrict__ y,
                                const int* __restrict__ idx,
                                float* __restrict__ out) {
    int row = blockIdx.x;
    int t   = threadIdx.x;
    int r   = idx[row];
    float gate = tanhf(y[r]);
    out[(size_t)row * TK_F + t] = X[(size_t)r * TK_F + t] * gate;
}

// ---- K4: A_pooled[i,j] = A[idx[i], idx[j]] --------------------------------
// 8 output rows per block; full idx staged once in 32KB LDS; coalesced writes.
__global__ void tk_apool_kernel(const float* __restrict__ A,
                                const int* __restrict__ idx,
                                float* __restrict__ out) {
    __shared__ int s_idx[TK_K];
    int t = threadIdx.x;                       // 1024 threads
    for (int k = t; k < TK_K; k += 1024) s_idx[k] = idx[k];
    __syncthreads();

    int rowblock = blockIdx.x * 8;
    for (int rr = 0; rr < 8; ++rr) {
        int row = rowblock + rr;
        const float* arow = A + (size_t)s_idx[row] * TK_N;
        float* orow = out + (size_t)row * TK_K;
        for (int j = t; j < TK_K; j += 1024)
            orow[j] = arow[s_idx[j]];          // gathered read, coalesced write
    }
}

// ---------------------------------------------------------------------------
extern "C" void kernel_launch(void* const* d_in, const int* in_sizes, int n_in,
                              void* d_out, int out_size, void* d_ws, size_t ws_size,
                              hipStream_t stream) {
    const float* X = (const float*)d_in[0];   // [N,F]
    const float* A = (const float*)d_in[1];   // [N,N]
    const float* p = (const float*)d_in[2];   // [F,1]

    float* out = (float*)d_out;
    float* Xp  = out;                          // [K,F] = 2,097,152 floats
    float* Ap  = out + (size_t)TK_K * TK_F;    // [K,K]

    // workspace: pnorm | y[N] | keys[N] | idx[K]   (~164 KB)
    float*    ws   = (float*)d_ws;
    float*    pn   = ws;
    float*    y    = ws + 64;
    unsigned* keys = (unsigned*)(ws + 64 + TK_N);
    int*      idx  = (int*)(ws + 64 + 2 * TK_N);

    tk_pnorm_kernel <<<1, TK_F, 0, stream>>>(p, pn);
    tk_scores_kernel<<<TK_N / 64, 128, 0, stream>>>(X, p, pn, y, keys);
    tk_select_kernel<<<1, 1024, 0, stream>>>(keys, idx);
    tk_xpool_kernel <<<TK_K, TK_F, 0, stream>>>(X, y, idx, Xp);
    tk_apool_kernel <<<TK_K / 8, 1024, 0, stream>>>(A, idx, Ap);
}